// ExpertChoiceRouting_44117904065241
// MI455X (gfx1250) — compile-verified
//
#include <hip/hip_runtime.h>
#include <hip/hip_bf16.h>

// Problem constants (from reference: B=8, S=4096, H=2048, E=64, CAPACITY=1.25)
#define T_TOK   32768      // B*S
#define H_DIM   2048
#define E_EXP   64
#define K_CAP   640        // int(1.25 * T / E)
#define KC      128        // K-chunk staged in LDS
#define LDSROW  132        // padded row stride (floats): 132 % 64 == 4 -> conflict-free b64 reads

typedef float v2f __attribute__((ext_vector_type(2)));
typedef float v8f __attribute__((ext_vector_type(8)));

// ---------------------------------------------------------------------------
// Kernel 1: fused router GEMM (fp32 WMMA 16x16x4) + softmax over experts.
// Grid: 256 blocks x 256 threads. Each block: 8 waves x 16 tokens = 128 tokens.
// Each wave computes a 16x64 logit tile as 4 WMMA accumulators over K=2048.
// ---------------------------------------------------------------------------
__global__ __launch_bounds__(256)
void router_gemm_softmax(const float* __restrict__ hs,
                         const float* __restrict__ W,
                         float* __restrict__ probs) {
    __shared__ float sW[E_EXP * LDSROW];   // 64 x 132 floats = 33 KB

    const int tid   = threadIdx.x;
    const int lane  = tid & 31;
    const int wv    = tid >> 5;            // wave id 0..7
    const int mrow  = lane & 15;           // M (token) / N (expert) sub-index
    const int khalf = lane >> 4;           // 0: K pair {0,1}, 1: K pair {2,3}
    const int tbase = blockIdx.x * 128 + wv * 16;

    v8f acc[4];
    #pragma unroll
    for (int nt = 0; nt < 4; ++nt) acc[nt] = (v8f)0.0f;

    const float* aRow = hs + (size_t)(tbase + mrow) * H_DIM + 2 * khalf;

    for (int kc = 0; kc < H_DIM / KC; ++kc) {
        __syncthreads();
        // Cooperative stage of router_weight chunk [64 x 128] into LDS.
        // 2048 float4s / 256 threads = 8 per thread, fully coalesced.
        #pragma unroll
        for (int f = tid; f < E_EXP * (KC / 4); f += 256) {
            const int row = f >> 5;        // expert
            const int c4  = f & 31;        // float4 index within chunk
            float4 v = *(const float4*)(W + (size_t)row * H_DIM + kc * KC + c4 * 4);
            *(float4*)&sW[row * LDSROW + c4 * 4] = v;
        }
        __syncthreads();

        const float* aP = aRow + kc * KC;
        #pragma unroll 4
        for (int i = 0; i < KC / 4; ++i) {
            // A fragment: lane holds hs[token = tbase+mrow][k + 2*khalf + {0,1}]
            const float2 af = *(const float2*)(aP + 4 * i);
            v2f a; a.x = af.x; a.y = af.y;
            #pragma unroll
            for (int nt = 0; nt < 4; ++nt) {
                // B fragment: lane holds W[expert = nt*16+mrow][k + 2*khalf + {0,1}]
                const float2 bf = *(const float2*)&sW[(nt * 16 + mrow) * LDSROW + 4 * i + 2 * khalf];
                v2f b; b.x = bf.x; b.y = bf.y;
                acc[nt] = __builtin_amdgcn_wmma_f32_16x16x4_f32(
                    false, a, false, b, (short)0, acc[nt], false, false);
            }
        }
    }

    // Softmax over experts (N axis): C layout -> VGPR r, lanes 0-15: M=r,
    // lanes 16-31: M=r+8; N = nt*16 + (lane&15). Reduce over 16-lane halves.
    #pragma unroll
    for (int r = 0; r < 8; ++r) {
        float v0 = acc[0][r], v1 = acc[1][r], v2 = acc[2][r], v3 = acc[3][r];
        float mx = fmaxf(fmaxf(v0, v1), fmaxf(v2, v3));
        #pragma unroll
        for (int off = 8; off >= 1; off >>= 1)
            mx = fmaxf(mx, __shfl_xor(mx, off, 32));
        float e0 = __expf(v0 - mx), e1 = __expf(v1 - mx);
        float e2 = __expf(v2 - mx), e3 = __expf(v3 - mx);
        float s = e0 + e1 + e2 + e3;
        #pragma unroll
        for (int off = 8; off >= 1; off >>= 1)
            s += __shfl_xor(s, off, 32);
        const float inv = 1.0f / s;
        const size_t rowOff = (size_t)(tbase + r + 8 * khalf) * E_EXP + mrow;
        probs[rowOff + 0 * 16] = e0 * inv;
        probs[rowOff + 1 * 16] = e1 * inv;
        probs[rowOff + 2 * 16] = e2 * inv;
        probs[rowOff + 3 * 16] = e3 * inv;
    }
}

// ---------------------------------------------------------------------------
// Kernel 2: per-expert exact top-K_CAP via 4-pass radix select on fp32 bits
// (monotone for positive probs), then deterministic in-order scatter with
// lowest-index tie-breaking. One block (256 threads) per expert; probs column
// is 128 KB and L2-resident.
// ---------------------------------------------------------------------------
__global__ __launch_bounds__(256)
void select_scatter(const float* __restrict__ probs,
                    float* __restrict__ dispatch) {
    const int e    = blockIdx.x;
    const int tid  = threadIdx.x;
    const int lane = tid & 31;
    const int wv   = tid >> 5;

    __shared__ unsigned hist[256];
    __shared__ unsigned sPrefix, sNeed, sTieBase;
    __shared__ unsigned waveCnt[8];

    unsigned prefix = 0u;
    unsigned need   = K_CAP;

    for (int pass = 3; pass >= 0; --pass) {
        const int shift = pass * 8;
        const unsigned maskHi = (pass == 3) ? 0u : (0xFFFFFFFFu << ((pass + 1) * 8));
        hist[tid] = 0u;
        __syncthreads();
        for (int t = tid; t < T_TOK; t += 256) {
            const unsigned u = __float_as_uint(probs[(size_t)t * E_EXP + e]);
            if ((u & maskHi) == (prefix & maskHi))
                atomicAdd(&hist[(u >> shift) & 0xFFu], 1u);
        }
        __syncthreads();
        if (tid == 0) {
            unsigned n = need;
            int b = 255;
            for (; b > 0; --b) {
                if (hist[b] >= n) break;
                n -= hist[b];
            }
            sPrefix = prefix | ((unsigned)b << shift);
            sNeed   = n;
        }
        __syncthreads();
        prefix = sPrefix;
        need   = sNeed;
        __syncthreads();
    }

    const unsigned theta    = prefix;   // bit pattern of k-th largest prob
    const unsigned needTies = need;     // #values == theta to accept (index order)
    if (tid == 0) sTieBase = 0u;
    __syncthreads();

    for (int b0 = 0; b0 < T_TOK; b0 += 256) {
        const int t = b0 + tid;
        const float p = probs[(size_t)t * E_EXP + e];
        const unsigned u = __float_as_uint(p);
        const bool gt = (u > theta);
        const bool eq = (u == theta);

        const unsigned long long bal = __ballot(eq);
        const unsigned mlo  = (unsigned)bal;
        const unsigned rank = __popc(mlo & ((1u << lane) - 1u));
        if (lane == 0) waveCnt[wv] = __popc(mlo);
        __syncthreads();

        unsigned waveOff = 0u, tot = 0u;
        #pragma unroll
        for (int w2 = 0; w2 < 8; ++w2) {
            const unsigned c = waveCnt[w2];
            if (w2 < wv) waveOff += c;
            tot += c;
        }
        const bool accept = gt || (eq && (sTieBase + waveOff + rank < needTies));
        dispatch[(size_t)t * E_EXP + e] = accept ? p : 0.0f;
        __syncthreads();
        if (tid == 0) sTieBase += tot;
        __syncthreads();
    }
}

// ---------------------------------------------------------------------------
// Kernel 3: combine_weights = dispatch / rowsum(dispatch) (0 when rowsum==0).
// ---------------------------------------------------------------------------
__global__ __launch_bounds__(256)
void combine_norm(const float* __restrict__ dispatch,
                  float* __restrict__ comb) {
    const int t = blockIdx.x * 256 + threadIdx.x;
    if (t >= T_TOK) return;
    const float4* dp = (const float4*)(dispatch + (size_t)t * E_EXP);
    float4 v[16];
    float s = 0.0f;
    #pragma unroll
    for (int i = 0; i < 16; ++i) {
        v[i] = dp[i];
        s += v[i].x + v[i].y + v[i].z + v[i].w;
    }
    const float inv = (s > 0.0f) ? (1.0f / s) : 0.0f;
    float4* cp = (float4*)(comb + (size_t)t * E_EXP);
    #pragma unroll
    for (int i = 0; i < 16; ++i) {
        cp[i] = make_float4(v[i].x * inv, v[i].y * inv, v[i].z * inv, v[i].w * inv);
    }
}

// ---------------------------------------------------------------------------
extern "C" void kernel_launch(void* const* d_in, const int* in_sizes, int n_in,
                              void* d_out, int out_size, void* d_ws, size_t ws_size,
                              hipStream_t stream) {
    (void)in_sizes; (void)n_in; (void)out_size; (void)d_ws; (void)ws_size;
    const float* hs = (const float*)d_in[0];   // [T, H] fp32
    const float* W  = (const float*)d_in[1];   // [E, H] fp32

    float* dispatch = (float*)d_out;                                // [T, E]
    float* comb     = (float*)d_out + (size_t)T_TOK * E_EXP;        // [T, E]
    float* probs    = (float*)d_out + 2 * (size_t)T_TOK * E_EXP;    // [T, E]

    router_gemm_softmax<<<T_TOK / 128, 256, 0, stream>>>(hs, W, probs);
    select_scatter<<<E_EXP, 256, 0, stream>>>(probs, dispatch);
    combine_norm<<<T_TOK / 256, 256, 0, stream>>>(dispatch, comb);
}